// Clustering_43877385896520
// MI455X (gfx1250) — compile-verified
//
#include <hip/hip_runtime.h>
#include <math.h>

typedef float v2f __attribute__((ext_vector_type(2)));
typedef float v8f __attribute__((ext_vector_type(8)));

#define B_   8
#define H_   8
#define L_   512
#define DK_  64
#define C_   5
#define HD_  512                 // H_*DK_
#define KBH  32768               // per-(b,h) stride: 512*64
#define KBATCH 262144            // per-b stride: 8*512*64
#define NOUT 16777216            // 8*8*512*512
#define MEANS_OFF 640            // ws float offset of means[5*512]

// ---------------------------------------------------------------------------
// Kernel 1: partial reductions for ck_proj[b][c] = sum_i K[b,i]*W_proj[i,c]
// grid (40, 16): bc = (b*5+c), 16 slices of the 262144-long reduction.
// ---------------------------------------------------------------------------
__global__ void proj_partial_kernel(const float* __restrict__ K,
                                    const float* __restrict__ W_proj,
                                    float* __restrict__ ws) {
    __shared__ float red[256];
    int bc = blockIdx.x;             // 0..39
    int s  = blockIdx.y;             // 0..15
    int b0 = bc / C_;
    int c  = bc % C_;
    const float* kp = K + (size_t)b0 * KBATCH;
    int i0 = s * (KBATCH / 16);
    float acc = 0.f;
    for (int i = i0 + threadIdx.x; i < i0 + KBATCH / 16; i += 256)
        acc += kp[i] * W_proj[(size_t)i * C_ + c];
    red[threadIdx.x] = acc;
    __syncthreads();
    for (int off = 128; off > 0; off >>= 1) {
        if (threadIdx.x < off) red[threadIdx.x] += red[threadIdx.x + off];
        __syncthreads();
    }
    if (threadIdx.x == 0) ws[bc * 16 + s] = red[0];
}

// ---------------------------------------------------------------------------
// Kernel 2: tiny stats (softmax, mu, sigma, loss, argmax) + means(5,512)
// single block of 256 threads.
// ---------------------------------------------------------------------------
__global__ void stats_kernel(const float* __restrict__ K,
                             const float* __restrict__ b_proj,
                             const float* __restrict__ W_q,
                             const float* __restrict__ b_q,
                             const float* __restrict__ W_k,
                             const float* __restrict__ b_k,
                             float* __restrict__ ws,
                             float* __restrict__ loss_out) {
    __shared__ int inds_s[B_];
    if (threadIdx.x == 0) {
        float ck[B_][C_];
        for (int b0 = 0; b0 < B_; b0++)
            for (int c = 0; c < C_; c++) {
                float a = b_proj[c];
                for (int s = 0; s < 16; s++) a += ws[(b0 * C_ + c) * 16 + s];
                ck[b0][c] = a;
            }
        float cq[B_][C_], cks[B_][C_];
        for (int b0 = 0; b0 < B_; b0++) {
            float lq[C_], lk[C_];
            for (int c = 0; c < C_; c++) {
                float aq = b_q[c], ak = b_k[c];
                for (int c2 = 0; c2 < C_; c2++) {
                    aq += ck[b0][c2] * W_q[c2 * C_ + c];
                    ak += ck[b0][c2] * W_k[c2 * C_ + c];
                }
                lq[c] = aq; lk[c] = ak;
            }
            float mq = lq[0], mk = lk[0];
            for (int c = 1; c < C_; c++) { mq = fmaxf(mq, lq[c]); mk = fmaxf(mk, lk[c]); }
            float sq = 0.f, sk = 0.f;
            for (int c = 0; c < C_; c++) {
                lq[c] = expf(lq[c] - mq); lk[c] = expf(lk[c] - mk);
                sq += lq[c]; sk += lk[c];
            }
            for (int c = 0; c < C_; c++) { cq[b0][c] = lq[c] / sq; cks[b0][c] = lk[c] / sk; }
        }
        float mu[C_], sigma[C_];
        for (int c = 0; c < C_; c++) {
            float m = 0.f;  for (int b0 = 0; b0 < B_; b0++) m += cq[b0][c];
            mu[c] = m / B_;
            float mk = 0.f; for (int b0 = 0; b0 < B_; b0++) mk += cks[b0][c];
            mk /= B_;
            float v = 0.f;
            for (int b0 = 0; b0 < B_; b0++) { float d = cks[b0][c] - mk; v += d * d; }
            float sd = sqrtf(v / (B_ - 1));
            sigma[c] = log1pf(expf(sd));           // softplus(std)
        }
        const float HALF_LOG_2PI = 0.9189385332046727f;
        float lpsum = 0.f;
        for (int b0 = 0; b0 < B_; b0++)
            for (int c = 0; c < C_; c++) {
                float z = (cks[b0][c] - mu[c]) / sigma[c];
                lpsum += -0.5f * z * z - logf(sigma[c]) - HALF_LOG_2PI;
            }
        float ce = 0.f;
        for (int b0 = 0; b0 < B_; b0++) {
            float mm = cq[b0][0];
            for (int c = 1; c < C_; c++) mm = fmaxf(mm, cq[b0][c]);
            float se = 0.f;
            for (int c = 0; c < C_; c++) se += expf(cq[b0][c] - mm);
            float lse = mm + logf(se);
            float e = 0.f;
            for (int c = 0; c < C_; c++) e += cq[b0][c] * (cq[b0][c] - lse);
            ce += -e;
        }
        ce /= B_;
        *loss_out = -(lpsum / (float)(B_ * C_)) + ce;
        for (int b0 = 0; b0 < B_; b0++) {
            int best = 0; float bv = cq[b0][0];
            for (int c = 1; c < C_; c++) if (cq[b0][c] > bv) { bv = cq[b0][c]; best = c; }
            inds_s[b0] = best;
        }
    }
    __syncthreads();
    // means[c][j] = (1/8) * sum_b (inds[b] != c+1) * K[b, 0:512]
    float* means = ws + MEANS_OFF;
    for (int idx = threadIdx.x; idx < C_ * HD_; idx += 256) {
        int c = idx / HD_, j = idx % HD_;
        float a = 0.f;
        for (int b0 = 0; b0 < B_; b0++)
            if (inds_s[b0] != c + 1) a += K[(size_t)b0 * KBATCH + j];
        means[idx] = a * 0.125f;
    }
}

// ---------------------------------------------------------------------------
// Kernel 3: main fused GEMM + cluster-max epilogue.
// grid (8 k-tiles, 4 q-tiles, 64 bh), 256 threads = 8 waves.
// Each wave: 16q x 64k tile via V_WMMA_F32_16X16X4_F32.
// ---------------------------------------------------------------------------
__global__ void __launch_bounds__(256, 2)
scores_kernel(const float* __restrict__ Q, const float* __restrict__ Kg,
              const float* __restrict__ ws, float* __restrict__ out) {
    __shared__ float means_s[C_ * HD_];   // 10 KB
    __shared__ float P_s[128 * 40];       // 20 KB
    const int bh = blockIdx.z;
    const int h  = bh & 7;
    const int qb = blockIdx.y * 128;
    const int kb = blockIdx.x * 64;
    const float* Qb = Q  + (size_t)bh * KBH;
    const float* Kb = Kg + (size_t)bh * KBH;
    const int tid = threadIdx.x;

    for (int i = tid; i < C_ * HD_; i += 256) means_s[i] = ws[MEANS_OFF + i];
    __syncthreads();

    // P[q][cp*8+m] = 0.125 * dot(Q[q,:], means[cp][m*64:(m+1)*64]); 2 threads per q.
    {
        int q = tid >> 1;
        int combo0 = (tid & 1) * 20;
        const float* qrow = Qb + (size_t)(qb + q) * DK_;
        float acc[20];
        int base[20];
        #pragma unroll
        for (int j = 0; j < 20; j++) {
            int combo = combo0 + j;
            base[j] = (combo >> 3) * HD_ + (combo & 7) * DK_;
            acc[j] = 0.f;
        }
        for (int d = 0; d < DK_; d++) {
            float qv = qrow[d];
            #pragma unroll
            for (int j = 0; j < 20; j++) acc[j] += qv * means_s[base[j] + d];
        }
        #pragma unroll
        for (int j = 0; j < 20; j++) P_s[q * 40 + combo0 + j] = acc[j] * 0.125f;
    }
    __syncthreads();

    // --- WMMA GEMM: scores_qk tile ---
    const int wave = tid >> 5, lane = tid & 31;
    const int row = lane & 15, half = lane >> 4;
    const int q0 = qb + wave * 16;
    const float* Ap  = Qb + (size_t)(q0 + row) * DK_ + 2 * half;
    const float* Bp0 = Kb + (size_t)(kb + row) * DK_ + 2 * half;
    const float* Bp1 = Bp0 + 16 * DK_;
    const float* Bp2 = Bp0 + 32 * DK_;
    const float* Bp3 = Bp0 + 48 * DK_;

    v8f acc[4];
    #pragma unroll
    for (int t = 0; t < 4; t++) acc[t] = (v8f)(0.f);

    #pragma unroll
    for (int kk = 0; kk < 16; kk++) {
        v2f a  = *(const v2f*)(Ap  + 4 * kk);
        v2f b0 = *(const v2f*)(Bp0 + 4 * kk);
        v2f b1 = *(const v2f*)(Bp1 + 4 * kk);
        v2f b2 = *(const v2f*)(Bp2 + 4 * kk);
        v2f b3 = *(const v2f*)(Bp3 + 4 * kk);
        acc[0] = __builtin_amdgcn_wmma_f32_16x16x4_f32(false, a, false, b0, (short)0, acc[0], false, false);
        acc[1] = __builtin_amdgcn_wmma_f32_16x16x4_f32(false, a, false, b1, (short)0, acc[1], false, false);
        acc[2] = __builtin_amdgcn_wmma_f32_16x16x4_f32(false, a, false, b2, (short)0, acc[2], false, false);
        acc[3] = __builtin_amdgcn_wmma_f32_16x16x4_f32(false, a, false, b3, (short)0, acc[3], false, false);
    }

    // --- epilogue: final = max(0.125*qk, max_c P[cp(h,c,k/8)][k%8]) ---
    const float scale = 0.125f;
    #pragma unroll
    for (int t = 0; t < 4; t++) {
        int k  = kb + t * 16 + row;          // N = lane%16
        int k8 = k >> 3, m = k & 7;
        int hb = h * 320 + k8;
        int cp0 = ((hb      ) >> 9) * 8 + m;
        int cp1 = ((hb +  64) >> 9) * 8 + m;
        int cp2 = ((hb + 128) >> 9) * 8 + m;
        int cp3 = ((hb + 192) >> 9) * 8 + m;
        int cp4 = ((hb + 256) >> 9) * 8 + m;
        #pragma unroll
        for (int i = 0; i < 8; i++) {
            int qloc = wave * 16 + i + 8 * half;   // M = i + 8*(lane/16)
            const float* Pq = P_s + qloc * 40;
            float mqc = Pq[cp0];
            mqc = fmaxf(mqc, Pq[cp1]);
            mqc = fmaxf(mqc, Pq[cp2]);
            mqc = fmaxf(mqc, Pq[cp3]);
            mqc = fmaxf(mqc, Pq[cp4]);
            float s = fmaxf(acc[t][i] * scale, mqc);
            out[((size_t)bh * L_ + (size_t)(qb + qloc)) * L_ + k] = s;
        }
    }
}

// ---------------------------------------------------------------------------
extern "C" void kernel_launch(void* const* d_in, const int* in_sizes, int n_in,
                              void* d_out, int out_size, void* d_ws, size_t ws_size,
                              hipStream_t stream) {
    (void)in_sizes; (void)n_in; (void)out_size; (void)ws_size;
    const float* Q      = (const float*)d_in[0];
    const float* K      = (const float*)d_in[1];
    const float* W_proj = (const float*)d_in[2];
    const float* b_proj = (const float*)d_in[3];
    const float* W_q    = (const float*)d_in[4];
    const float* b_q    = (const float*)d_in[5];
    const float* W_k    = (const float*)d_in[6];
    const float* b_k    = (const float*)d_in[7];
    float* out = (float*)d_out;
    float* ws  = (float*)d_ws;

    proj_partial_kernel<<<dim3(40, 16), 256, 0, stream>>>(K, W_proj, ws);
    stats_kernel<<<1, 256, 0, stream>>>(K, b_proj, W_q, b_q, W_k, b_k, ws, out + NOUT);
    scores_kernel<<<dim3(8, 4, 64), 256, 0, stream>>>(Q, K, ws, out);
}